// SIRNet_3925600109176
// MI455X (gfx1250) — compile-verified
//
#include <hip/hip_runtime.h>
#include <stdint.h>

// ---------------------------------------------------------------------------
// SIR recurrence, T=2000 sequential, B=8192 parallel. HBM-bound (~0.5 FLOP/B,
// 655 MB total -> ~28 us roofline at 23.3 TB/s).
//   - TDM (tensor_load_to_lds) stages X tiles into LDS, 8-deep per wave,
//     tracked with TENSORcnt -> ~27 MB of DMA in flight device-wide (above
//     the ~14-23 MB latency-BW product of HBM at 23.3 TB/s).
//   - One wave (32 lanes) per block, one batch column per lane, no barriers.
//   - b(t) (state-independent) computed in a separate ILP-rich pass per tile;
//     the short dependent recurrence chain follows.
//   - Per-lane non-temporal stores: each wave-timestep writes 3 full 128B
//     lines (hiddens) + 1 full line (outputs); TDM stores would add in-order
//     TENSORcnt coupling that throttles the dominant read stream for no
//     coalescing gain.
// ---------------------------------------------------------------------------

typedef __attribute__((ext_vector_type(4))) unsigned int u32x4;
typedef __attribute__((ext_vector_type(4))) int          i32x4;
typedef __attribute__((ext_vector_type(8))) int          i32x8;

#define SIR_I0 5.6e-06f

constexpr int T_STEPS = 2000;
constexpr int B_COLS  = 8192;
constexpr int COLS    = 32;              // columns per block (1 wave32)
constexpr int ROWF    = COLS * 6;        // 192 floats of X per timestep per block
constexpr int TT      = 20;              // timesteps per TDM tile (2000 = 100*20)
constexpr int NBUF    = 8;               // LDS tile buffers (pipeline depth)
constexpr int NTILES  = T_STEPS / TT;    // 100
constexpr int BUF_FLOATS = TT * ROWF;    // 3840 floats = 15360 B
constexpr int BUF_BYTES  = BUF_FLOATS * 4;
// Total LDS: 8 * 15360 B = 122880 B (CDNA5: up to 320 KB per workgroup)

// ---- TDM issue: 2D tile (tile_dim0 elems contiguous, tile_dim1 rows, row
// stride tensor_dim0_stride elems) from global -> LDS. Descriptor per ISA §8.
__device__ __forceinline__ void tdm_load_2d(uint32_t lds_byte_addr,
                                            const float* gptr,
                                            uint32_t tile0_elems,
                                            uint32_t tile1_rows,
                                            uint64_t stride0_elems) {
    const uint64_t ga = (uint64_t)(uintptr_t)gptr;

    // D# group 0: [1:0]=count=1, [63:32]=lds_addr, [120:64]=global_addr,
    // [127:126]=type=2.
    u32x4 g0;
    g0[0] = 1u;
    g0[1] = lds_byte_addr;
    g0[2] = (uint32_t)(ga & 0xFFFFFFFFu);
    g0[3] = (uint32_t)((ga >> 32) & 0x01FFFFFFu) | (2u << 30);

    // D# group 1: data_size=4B; generous tensor dims (tiles never OOB);
    // tile dims + dim0 stride; dim1 stride unused for 2D tiles.
    const uint32_t td0 = 0x40000000u;
    const uint32_t td1 = 0x40000000u;
    i32x8 g1;
    g1[0] = (int)(2u << 16);                                   // data_size=2 (4B)
    g1[1] = (int)((td0 & 0xFFFFu) << 16);                      // tensor_dim0[15:0]
    g1[2] = (int)(((td0 >> 16) & 0xFFFFu) | ((td1 & 0xFFFFu) << 16));
    g1[3] = (int)(((td1 >> 16) & 0xFFFFu) | ((tile0_elems & 0xFFFFu) << 16));
    g1[4] = (int)(tile1_rows & 0xFFFFu);                       // tile_dim1, tile_dim2=0
    g1[5] = (int)(uint32_t)(stride0_elems & 0xFFFFFFFFu);      // dim0_stride[31:0]
    g1[6] = (int)(uint32_t)((stride0_elems >> 32) & 0xFFFFu);  // dim0_stride[47:32]
    g1[7] = 0;

    i32x4 z4 = {0, 0, 0, 0};
#if __has_include(<hip/amd_detail/amd_gfx1250_TDM.h>)
    // therock / clang-23 toolchain: 6-arg builtin
    i32x8 z8 = {0, 0, 0, 0, 0, 0, 0, 0};
    __builtin_amdgcn_tensor_load_to_lds(g0, g1, z4, z4, z8, 0);
#else
    // ROCm 7.2 / clang-22 toolchain: 5-arg builtin
    __builtin_amdgcn_tensor_load_to_lds(g0, g1, z4, z4, 0);
#endif
    asm volatile("" ::: "memory");
}

#define WAIT_TENSORCNT(n)                              \
    do {                                               \
        __builtin_amdgcn_s_wait_tensorcnt(n);          \
        asm volatile("" ::: "memory");                 \
    } while (0)

__device__ __forceinline__ void wait_ds0() {
#if __has_builtin(__builtin_amdgcn_s_wait_dscnt)
    __builtin_amdgcn_s_wait_dscnt(0);
    asm volatile("" ::: "memory");
#else
    asm volatile("s_wait_dscnt 0" ::: "memory");
#endif
}

__global__ void __launch_bounds__(COLS)
sir_tdm_kernel(const float* __restrict__ X,
               const float* __restrict__ w_b,    // 6
               const float* __restrict__ kparam, // 1
               const float* __restrict__ w_h2o,  // 3
               float* __restrict__ out_h,        // [T,B,3]
               float* __restrict__ out_o)        // [T,B,1]
{
    __shared__ float xbuf[NBUF * BUF_FLOATS];   // 122880 B

    const int lane = threadIdx.x;               // 0..31
    const int c    = blockIdx.x * COLS + lane;  // batch column

    // Uniform weights (scalar loads).
    const float w0 = w_b[0], w1 = w_b[1], w2 = w_b[2];
    const float w3 = w_b[3], w4 = w_b[4], w5 = w_b[5];
    const float kk = kparam[0];
    const float h0 = w_h2o[0], h1 = w_h2o[1], h2 = w_h2o[2];

    // LDS byte offset of xbuf: low 32 bits of the generic (flat) address.
    const uint32_t lds0 = (uint32_t)(uintptr_t)(void*)xbuf;

    const float*   gbase        = X + (size_t)blockIdx.x * COLS * 6;
    const size_t   tile_gstride = (size_t)TT * B_COLS * 6;   // floats per tile step
    const uint64_t row_stride   = (uint64_t)B_COLS * 6;      // 49152 elems

    // Prologue: fill the pipeline with NBUF tile loads.
#pragma unroll
    for (int i = 0; i < NBUF; ++i) {
        tdm_load_2d(lds0 + (uint32_t)i * BUF_BYTES,
                    gbase + (size_t)i * tile_gstride,
                    ROWF, TT, row_stride);
    }

    float p0 = SIR_I0, p1 = 0.0f, p2 = 1.0f - SIR_I0;

    for (int tile = 0; tile < NTILES; ++tile) {
        // Ensure tile's DMA completed. In-order TENSORcnt: <= NBUF-1 leaves
        // only the (up to) NBUF-1 newer loads outstanding. Tail tiles have
        // fewer outstanding; drain fully there.
        if (tile < NTILES - NBUF + 1) {
            WAIT_TENSORCNT(NBUF - 1);
        } else {
            WAIT_TENSORCNT(0);
        }

        const int    buf = tile & (NBUF - 1);
        const float* lx  = xbuf + buf * BUF_FLOATS + lane * 6;

        // Phase 1: state-independent b(t) for the whole tile (ILP-rich,
        // pipelined ds_loads, no dependence chain).
        float barr[TT];
#pragma unroll
        for (int j = 0; j < TT; ++j) {
            const float* xp = lx + j * ROWF;
            float s = xp[0] * w0 + xp[1] * w1 + xp[2] * w2
                    + xp[3] * w3 + xp[4] * w4 + xp[5] * w5;
            barr[j] = fmaxf(s, 0.0f);
        }

        // All LDS reads of this buffer have drained; recycle it.
        wait_ds0();
        if (tile + NBUF < NTILES) {
            tdm_load_2d(lds0 + (uint32_t)buf * BUF_BYTES,
                        gbase + (size_t)(tile + NBUF) * tile_gstride,
                        ROWF, TT, row_stride);
        }

        // Phase 2: sequential SIR recurrence + streaming NT stores.
        const int t0 = tile * TT;
#pragma unroll
        for (int j = 0; j < TT; ++j) {
            const float b    = barr[j];
            const float drdt = kk * p0;
            const float n0   = fmaf(p0 * b, p2, p0 - drdt); // p0 + p0*b*p2 - k*p0
            const float n1   = p1 + drdt;
            const float n2   = 1.0f - p0 - p1;
            const float o    = n0 * h0 + n1 * h1 + n2 * h2;

            const size_t idx = (size_t)(t0 + j) * B_COLS + c;
            __builtin_nontemporal_store(n0, out_h + idx * 3 + 0);
            __builtin_nontemporal_store(n1, out_h + idx * 3 + 1);
            __builtin_nontemporal_store(n2, out_h + idx * 3 + 2);
            __builtin_nontemporal_store(o,  out_o + idx);

            p0 = n0; p1 = n1; p2 = n2;
        }
    }
}

extern "C" void kernel_launch(void* const* d_in, const int* in_sizes, int n_in,
                              void* d_out, int out_size, void* d_ws, size_t ws_size,
                              hipStream_t stream) {
    (void)in_sizes; (void)n_in; (void)out_size; (void)d_ws; (void)ws_size;

    const float* X     = (const float*)d_in[0];  // [T,B,6]
    const float* w_b   = (const float*)d_in[1];  // [1,6]
    const float* kpar  = (const float*)d_in[2];  // [1,1]
    const float* w_h2o = (const float*)d_in[3];  // [1,3]

    float* out   = (float*)d_out;
    float* out_h = out;                                       // [T,B,3]
    float* out_o = out + (size_t)T_STEPS * B_COLS * 3;        // [T,B,1]

    dim3 grid(B_COLS / COLS);  // 256 blocks
    dim3 block(COLS);          // 32 threads = 1 wave32
    sir_tdm_kernel<<<grid, block, 0, stream>>>(X, w_b, kpar, w_h2o, out_h, out_o);
}